// RGAT_59622736003344
// MI455X (gfx1250) — compile-verified
//
#include <hip/hip_runtime.h>
#include <hip/hip_bf16.h>
#include <stdint.h>

// Problem constants (match reference)
#define R_   8
#define IN_  128
#define H_   2
#define OUT_ 256
#define HO_  512

typedef __attribute__((ext_vector_type(2))) float v2f;
typedef __attribute__((ext_vector_type(8))) float v8f;
typedef __attribute__((ext_vector_type(4))) int   v4i;

// b128 async-copy operand types: pointer to 16B vector in AS1 (global) / AS3 (LDS)
typedef __attribute__((address_space(1))) v4i as1_v4i;
typedef __attribute__((address_space(3))) v4i as3_v4i;

#if __has_builtin(__builtin_amdgcn_global_load_async_to_lds_b128)
#define HAVE_ASYNC_LDS 1
#else
#define HAVE_ASYNC_LDS 0
#endif

__device__ __forceinline__ void waitAsyncAll() {
#if HAVE_ASYNC_LDS
#if __has_builtin(__builtin_amdgcn_s_wait_asynccnt)
  __builtin_amdgcn_s_wait_asynccnt(0);
#else
  asm volatile("s_wait_asynccnt 0" ::: "memory");
#endif
#endif
}

__device__ __forceinline__ void atomicAddF(float* p, float v) {
  unsafeAtomicAdd(p, v);  // native global_atomic_add_f32
}

// float atomic max via sign-split integer atomics (init value must be -inf)
__device__ __forceinline__ void atomicMaxF(float* addr, float v) {
  if (v >= 0.f) atomicMax((int*)addr, __float_as_int(v));
  else          atomicMin((unsigned int*)addr, (unsigned int)__float_as_int(v));
}

// ---------------------------------------------------------------------------
// Kernel 0: init out=bias broadcast, amax=-inf, denom=0
// ---------------------------------------------------------------------------
__global__ __launch_bounds__(256) void rgat_init(
    float* __restrict__ out, const float* __restrict__ bias,
    float* __restrict__ amax, float* __restrict__ denom,
    int totalOut, int NH) {
  int i = blockIdx.x * 256 + threadIdx.x;
  if (i < totalOut) out[i] = bias[i & (HO_ - 1)];
  if (i < NH) { amax[i] = -__builtin_inff(); denom[i] = 0.f; }
}

// ---------------------------------------------------------------------------
// Kernel 1: xw[r] = x @ W[r]   (f32 WMMA 16x16x4)
// block = 256 threads (8 wave32); block tile = 128 rows x 64 cols; K = 128
// ---------------------------------------------------------------------------
#define MT 128
#define NT 64
#define WSTRIDE 68  // padded LDS row stride (floats); 68*4 bytes = 16B aligned

__global__ __launch_bounds__(256) void rgat_xw_wmma(
    const float* __restrict__ x,   // [N, 128]
    const float* __restrict__ w,   // [R, 128, 512]
    float* __restrict__ xw,        // [R, N, 512]
    int nNodes) {
  __shared__ float lds_w[IN_ * WSTRIDE];  // ~34.8 KB

  const int r    = blockIdx.z;
  const int row0 = blockIdx.x * MT;
  const int col0 = blockIdx.y * NT;
  const int tid  = threadIdx.x;

  // Stage W[r][0:128, col0:col0+64] into LDS (row stride WSTRIDE).
  const float* wr = w + (size_t)r * IN_ * HO_ + col0;
#pragma unroll
  for (int i = 0; i < 8; ++i) {
    int idx = tid + i * 256;        // 2048 float4 chunks: 128 rows * 16 chunks
    int k   = idx >> 4;
    int c4  = (idx & 15) * 4;
#if HAVE_ASYNC_LDS
    __builtin_amdgcn_global_load_async_to_lds_b128(
        (as1_v4i*)(wr + (size_t)k * HO_ + c4),
        (as3_v4i*)&lds_w[k * WSTRIDE + c4], 0, 0);
#else
    *(float4*)&lds_w[k * WSTRIDE + c4] = *(const float4*)(wr + (size_t)k * HO_ + c4);
#endif
  }
  waitAsyncAll();
  __syncthreads();

  const int wave = tid >> 5;       // wave32
  const int lane = tid & 31;
  const int half = lane >> 4;      // K-pair select for A/B fragments
  const int mr   = lane & 15;      // M row (A) / N col (B)

  const int  grow  = row0 + wave * 16 + mr;
  const bool valid = grow < nNodes;
  const float* xrow = x + (size_t)(valid ? grow : 0) * IN_;

  v8f acc[4] = {};                 // 4 tiles of 16x16 => 16 rows x 64 cols / wave

#pragma unroll 4
  for (int k0 = 0; k0 < IN_; k0 += 4) {
    const int ka = k0 + 2 * half;
    v2f a = *(const v2f*)(xrow + ka);            // A[16x4] fragment
#pragma unroll
    for (int t = 0; t < 4; ++t) {
      const float* wb = &lds_w[ka * WSTRIDE + t * 16 + mr];
      v2f b = { wb[0], wb[WSTRIDE] };            // B[4x16] fragment
#if __has_builtin(__builtin_amdgcn_wmma_f32_16x16x4_f32)
      acc[t] = __builtin_amdgcn_wmma_f32_16x16x4_f32(
          false, a, false, b, (short)0, acc[t], false, false);
#else
      // degenerate fallback (should not trigger per toolchain probe)
#pragma unroll
      for (int j = 0; j < 8; ++j) acc[t][j] += a[0] * b[0] + a[1] * b[1];
#endif
    }
  }

  // C layout: lane holds C[8*half + j][t*16 + mr] in acc[t][j]
  float* obase = xw + (size_t)r * nNodes * HO_;
#pragma unroll
  for (int t = 0; t < 4; ++t) {
#pragma unroll
    for (int j = 0; j < 8; ++j) {
      int rr = row0 + wave * 16 + half * 8 + j;
      if (rr < nNodes) obase[(size_t)rr * HO_ + col0 + t * 16 + mr] = acc[t][j];
    }
  }
}

// ---------------------------------------------------------------------------
// Kernel 2: qn/kn[r,n,h] = xw[r,n,:] . q/k[:,h]
// ---------------------------------------------------------------------------
__global__ __launch_bounds__(256) void rgat_qk(
    const float* __restrict__ xw, const float* __restrict__ qm,
    const float* __restrict__ km, float* __restrict__ qn,
    float* __restrict__ kn, int totalRN) {
  __shared__ float sq[HO_ * H_];
  __shared__ float sk[HO_ * H_];
  for (int i = threadIdx.x; i < HO_ * H_; i += 256) { sq[i] = qm[i]; sk[i] = km[i]; }
  __syncthreads();

  int idx = blockIdx.x * 256 + threadIdx.x;
  if (idx >= totalRN) return;
  const float* row = xw + (size_t)idx * HO_;
  float a0 = 0.f, a1 = 0.f, a2 = 0.f, a3 = 0.f;
#pragma unroll 4
  for (int o = 0; o < HO_; o += 4) {
    float4 v = *(const float4*)(row + o);
    a0 += v.x * sq[(o + 0) * 2] + v.y * sq[(o + 1) * 2] + v.z * sq[(o + 2) * 2] + v.w * sq[(o + 3) * 2];
    a1 += v.x * sq[(o + 0) * 2 + 1] + v.y * sq[(o + 1) * 2 + 1] + v.z * sq[(o + 2) * 2 + 1] + v.w * sq[(o + 3) * 2 + 1];
    a2 += v.x * sk[(o + 0) * 2] + v.y * sk[(o + 1) * 2] + v.z * sk[(o + 2) * 2] + v.w * sk[(o + 3) * 2];
    a3 += v.x * sk[(o + 0) * 2 + 1] + v.y * sk[(o + 1) * 2 + 1] + v.z * sk[(o + 2) * 2 + 1] + v.w * sk[(o + 3) * 2 + 1];
  }
  qn[(size_t)idx * 2]     = a0;
  qn[(size_t)idx * 2 + 1] = a1;
  kn[(size_t)idx * 2]     = a2;
  kn[(size_t)idx * 2 + 1] = a3;
}

// ---------------------------------------------------------------------------
// Kernel 3: per-edge alpha = leaky_relu(q_i + k_j); segment max over dst
// ---------------------------------------------------------------------------
__global__ __launch_bounds__(256) void rgat_alpha(
    const int* __restrict__ ei, const int* __restrict__ el,
    const float* __restrict__ qn, const float* __restrict__ kn,
    float* __restrict__ ealpha, float* __restrict__ amax, int E_, int nNodes) {
  int e = blockIdx.x * 256 + threadIdx.x;
  if (e >= E_) return;
  int s = ei[e], d = ei[E_ + e], r = el[e];
  size_t qi = ((size_t)r * nNodes + d) * 2;
  size_t kj = ((size_t)r * nNodes + s) * 2;
  float a0 = qn[qi] + kn[kj];
  float a1 = qn[qi + 1] + kn[kj + 1];
  a0 = (a0 > 0.f) ? a0 : 0.2f * a0;
  a1 = (a1 > 0.f) ? a1 : 0.2f * a1;
  ealpha[(size_t)e * 2]     = a0;
  ealpha[(size_t)e * 2 + 1] = a1;
  atomicMaxF(&amax[d * 2], a0);
  atomicMaxF(&amax[d * 2 + 1], a1);
}

// ---------------------------------------------------------------------------
// Kernel 4: ex = exp(alpha - amax[dst]); segment sum into denom
// ---------------------------------------------------------------------------
__global__ __launch_bounds__(256) void rgat_expden(
    const int* __restrict__ ei, const float* __restrict__ amax,
    float* __restrict__ ealpha, float* __restrict__ denom, int E_) {
  int e = blockIdx.x * 256 + threadIdx.x;
  if (e >= E_) return;
  int d = ei[E_ + e];
  float m0 = amax[d * 2];     if (m0 == -__builtin_inff()) m0 = 0.f;
  float m1 = amax[d * 2 + 1]; if (m1 == -__builtin_inff()) m1 = 0.f;
  float ex0 = __expf(ealpha[(size_t)e * 2] - m0);
  float ex1 = __expf(ealpha[(size_t)e * 2 + 1] - m1);
  ealpha[(size_t)e * 2]     = ex0;
  ealpha[(size_t)e * 2 + 1] = ex1;
  atomicAddF(&denom[d * 2], ex0);
  atomicAddF(&denom[d * 2 + 1], ex1);
}

// ---------------------------------------------------------------------------
// Kernel 5: scatter messages: out[dst] += (ex/denom) * xw[r, src]
// one wave32 per edge; 512 floats per edge; 16 native f32 atomics per lane
// ---------------------------------------------------------------------------
__global__ __launch_bounds__(256) void rgat_scatter(
    const int* __restrict__ ei, const int* __restrict__ el,
    const float* __restrict__ ealpha, const float* __restrict__ denom,
    const float* __restrict__ xw, float* __restrict__ out, int E_, int nNodes) {
  int e    = blockIdx.x * 8 + (threadIdx.x >> 5);
  int lane = threadIdx.x & 31;
  if (e >= E_) return;
  int s = ei[e], d = ei[E_ + e], r = el[e];
  float an0 = ealpha[(size_t)e * 2]     / (denom[d * 2] + 1e-16f);
  float an1 = ealpha[(size_t)e * 2 + 1] / (denom[d * 2 + 1] + 1e-16f);
  const float* srow = xw + ((size_t)r * nNodes + s) * HO_;
  float* drow = out + (size_t)d * HO_;
  __builtin_prefetch(srow + lane * 4, 0, 0);  // global_prefetch_b8
#pragma unroll
  for (int i = 0; i < 4; ++i) {
    int o = lane * 4 + i * 128;                  // 0..508, unique 4-aligned
    float4 v = *(const float4*)(srow + o);
    float an = (o < OUT_) ? an0 : an1;           // head = o / 256
    atomicAddF(drow + o + 0, an * v.x);
    atomicAddF(drow + o + 1, an * v.y);
    atomicAddF(drow + o + 2, an * v.z);
    atomicAddF(drow + o + 3, an * v.w);
  }
}

// ---------------------------------------------------------------------------
extern "C" void kernel_launch(void* const* d_in, const int* in_sizes, int n_in,
                              void* d_out, int out_size, void* d_ws, size_t ws_size,
                              hipStream_t stream) {
  const float* x    = (const float*)d_in[0];
  const int*   ei   = (const int*)d_in[1];
  const int*   el   = (const int*)d_in[2];
  const float* w    = (const float*)d_in[3];
  const float* qm   = (const float*)d_in[4];
  const float* km   = (const float*)d_in[5];
  const float* bias = (const float*)d_in[6];
  float* out = (float*)d_out;

  const int N_ = in_sizes[0] / IN_;   // 10000
  const int E_ = in_sizes[2];         // 160000

  // Workspace carve (floats)
  float* ws     = (float*)d_ws;
  float* xw     = ws;                                  // R*N*HO
  float* qn     = xw + (size_t)R_ * N_ * HO_;          // R*N*H
  float* kn     = qn + (size_t)R_ * N_ * H_;           // R*N*H
  float* amax   = kn + (size_t)R_ * N_ * H_;           // N*H
  float* denom  = amax + (size_t)N_ * H_;              // N*H
  float* ealpha = denom + (size_t)N_ * H_;             // E*H

  const int totalOut = N_ * HO_;
  const int NH       = N_ * H_;
  rgat_init<<<(totalOut + 255) / 256, 256, 0, stream>>>(out, bias, amax, denom,
                                                        totalOut, NH);

  dim3 gg((N_ + MT - 1) / MT, HO_ / NT, R_);
  rgat_xw_wmma<<<gg, 256, 0, stream>>>(x, w, xw, N_);

  const int totalRN = R_ * N_;
  rgat_qk<<<(totalRN + 255) / 256, 256, 0, stream>>>(xw, qm, km, qn, kn, totalRN);

  rgat_alpha<<<(E_ + 255) / 256, 256, 0, stream>>>(ei, el, qn, kn, ealpha, amax,
                                                   E_, N_);
  rgat_expden<<<(E_ + 255) / 256, 256, 0, stream>>>(ei, amax, ealpha, denom, E_);
  rgat_scatter<<<(E_ + 7) / 8, 256, 0, stream>>>(ei, el, ealpha, denom, xw, out,
                                                 E_, N_);
}